// TELightGCN_1486058684585
// MI455X (gfx1250) — compile-verified
//
#include <hip/hip_runtime.h>

#define N_USERS  100000
#define N_ITEMS  50000
#define N_TOPICS 1000
#define N_NODES  (N_USERS + N_ITEMS + N_TOPICS)
#define DIM      64
#define NNZ      4000000
#define BATCH    16384

typedef __attribute__((ext_vector_type(2))) float v2f;
typedef __attribute__((ext_vector_type(8))) float v8f;

// ---------------------------------------------------------------------------
// Build all_emb = concat(user_w, item_w, topic_w) into both acc and emb0.
// Pure streaming copy, float4 vectorized, grid-stride.
// ---------------------------------------------------------------------------
__global__ void lgcn_init_concat(const float* __restrict__ uw,
                                 const float* __restrict__ iw,
                                 const float* __restrict__ tw,
                                 float* __restrict__ acc,
                                 float* __restrict__ emb) {
  const long total4 = (long)N_NODES * (DIM / 4);
  for (long i = blockIdx.x * (long)blockDim.x + threadIdx.x; i < total4;
       i += (long)gridDim.x * blockDim.x) {
    long node = i / (DIM / 4);
    long off  = i % (DIM / 4);
    float4 v;
    if (node < N_USERS)
      v = ((const float4*)uw)[node * (DIM / 4) + off];
    else if (node < N_USERS + N_ITEMS)
      v = ((const float4*)iw)[(node - N_USERS) * (DIM / 4) + off];
    else
      v = ((const float4*)tw)[(node - N_USERS - N_ITEMS) * (DIM / 4) + off];
    ((float4*)acc)[i] = v;
    ((float4*)emb)[i] = v;
  }
}

// ---------------------------------------------------------------------------
// COO SpMM: y[row[e]] += vals[e] * x[col[e]]  (64-dim rows).
// 16 lanes per edge, float4 per lane -> one coalesced 256B gather per edge,
// scatter via global_atomic_add_f32 (RMW resolved in L2; the 38.7MB node
// table is L2-resident on MI455X's 192MB L2).
// ---------------------------------------------------------------------------
__global__ void lgcn_spmm_atomic(const float* __restrict__ vals,
                                 const int* __restrict__ row,
                                 const int* __restrict__ col,
                                 const float* __restrict__ x,
                                 float* __restrict__ y) {
  long gid = blockIdx.x * (long)blockDim.x + threadIdx.x;
  long e   = gid >> 4;
  int  seg = (int)(gid & 15);
  if (e >= NNZ) return;
  // keep the streaming edge list ahead of us (emits global_prefetch_b8)
  __builtin_prefetch(col + e + 4096, 0, 3);
  float val = vals[e];
  long  r   = row[e];
  long  c   = col[e];
  float4 xv = ((const float4*)(x + c * DIM))[seg];
  float* yp = y + r * DIM + (long)seg * 4;
  atomicAdd(yp + 0, val * xv.x);
  atomicAdd(yp + 1, val * xv.y);
  atomicAdd(yp + 2, val * xv.z);
  atomicAdd(yp + 3, val * xv.w);
}

// ---------------------------------------------------------------------------
// acc += e   (float4, grid-stride)
// ---------------------------------------------------------------------------
__global__ void lgcn_accum(float* __restrict__ acc, const float* __restrict__ e) {
  const long total4 = (long)N_NODES * (DIM / 4);
  for (long i = blockIdx.x * (long)blockDim.x + threadIdx.x; i < total4;
       i += (long)gridDim.x * blockDim.x) {
    float4 a = ((const float4*)acc)[i];
    float4 b = ((const float4*)e)[i];
    a.x += b.x; a.y += b.y; a.z += b.z; a.w += b.w;
    ((float4*)acc)[i] = a;
  }
}

// ---------------------------------------------------------------------------
// Batched dot products via WMMA: one wave handles 16 (user,item) pairs.
// S = U(16x64) @ I^T(64x16) built from 16x V_WMMA_F32_16X16X4_F32 (K chunks
// of 4); result[b] = S[b][b] / 16 (folds the two 1/(LAYERS+1) scalings).
//
// Operand layout (ISA 7.12.2, 32-bit A 16x4 / B 4x16): lanes 0-15 carry
// K=(0,1) in VGPR(0,1), lanes 16-31 carry K=(2,3)  ->  per lane load
// row[4k + 2*(lane>=16) + {0,1}] for both A (users) and B (items).
// C/D diag: lanes 0-7 VGPR=lane hold S[m][m], m=0..7; lanes 24-31
// VGPR=lane-24 hold S[m][m], m=lane-16.
// ---------------------------------------------------------------------------
__global__ void lgcn_dot_wmma(const float* __restrict__ acc,
                              const int* __restrict__ users,
                              const int* __restrict__ items,
                              float* __restrict__ out) {
  const int wavesPerBlock = blockDim.x >> 5;
  int wave = blockIdx.x * wavesPerBlock + (threadIdx.x >> 5);
  int lane = threadIdx.x & 31;
  if (wave >= BATCH / 16) return;   // wave-uniform; EXEC stays all-ones

  int half = lane >> 4;             // 0: K pair (0,1), 1: K pair (2,3)
  int m16  = lane & 15;
  int u  = users[wave * 16 + m16];
  int it = items[wave * 16 + m16];
  const float* urow = acc + (size_t)u * DIM;
  const float* irow = acc + ((size_t)N_USERS + (size_t)it) * DIM;

  v8f c = {};
#pragma unroll
  for (int k = 0; k < 16; ++k) {
    int d = 4 * k + 2 * half;
    v2f a = { urow[d], urow[d + 1] };
    v2f b = { irow[d], irow[d + 1] };
    c = __builtin_amdgcn_wmma_f32_16x16x4_f32(
        /*neg_a=*/false, a, /*neg_b=*/false, b,
        /*c_mod=*/(short)0, c, /*reuse_a=*/false, /*reuse_b=*/false);
  }

  int  v      = lane & 7;
  float dval  = c[v];
  bool active = (lane < 8) || (lane >= 24);
  int  m      = (lane < 8) ? lane : (lane - 16);
  if (active) out[wave * 16 + m] = dval * 0.0625f;  // (1/4)*(1/4) scaling
}

// ---------------------------------------------------------------------------
extern "C" void kernel_launch(void* const* d_in, const int* in_sizes, int n_in,
                              void* d_out, int out_size, void* d_ws, size_t ws_size,
                              hipStream_t stream) {
  const float* user_w  = (const float*)d_in[0];
  const float* item_w  = (const float*)d_in[1];
  const float* topic_w = (const float*)d_in[2];
  const float* vals    = (const float*)d_in[3];
  const int*   row     = (const int*)d_in[4];
  const int*   col     = (const int*)d_in[5];
  const int*   users   = (const int*)d_in[6];
  const int*   items   = (const int*)d_in[7];
  float*       out     = (float*)d_out;

  const size_t nnd = (size_t)N_NODES * DIM;     // 9,664,000 floats
  float* acc  = (float*)d_ws;                   // [nnd]
  float* embA = acc + nnd;                      // [nnd]
  float* embB = embA + nnd;                     // [nnd]  (~116 MB total)

  lgcn_init_concat<<<2048, 256, 0, stream>>>(user_w, item_w, topic_w, acc, embA);

  float* x = embA;
  float* y = embB;
  const int spmmBlocks = (int)(((long)NNZ * 16 + 255) / 256);  // 250,000
  for (int l = 0; l < 3; ++l) {
    hipMemsetAsync(y, 0, nnd * sizeof(float), stream);
    lgcn_spmm_atomic<<<spmmBlocks, 256, 0, stream>>>(vals, row, col, x, y);
    lgcn_accum<<<2048, 256, 0, stream>>>(acc, y);
    float* t = x; x = y; y = t;
  }

  // 1024 tiles of 16 pairs; 4 waves/block -> 256 blocks, exact fit.
  lgcn_dot_wmma<<<(BATCH / 16) / 4, 128, 0, stream>>>(acc, users, items, out);
}